// SimpleSurrogate_80625126081179
// MI455X (gfx1250) — compile-verified
//
#include <hip/hip_runtime.h>
#include <hip/hip_bf16.h>

typedef float v2f __attribute__((ext_vector_type(2)));
typedef float v8f __attribute__((ext_vector_type(8)));

#define H 128
#define WPAD 132   // LDS row pitch (floats) to avoid bank conflicts

// ---------------- degree / dinv ----------------
__global__ void deg_kernel(const int* __restrict__ dst, float* __restrict__ deg, int E) {
    int e = blockIdx.x * blockDim.x + threadIdx.x;
    if (e < E) atomicAdd(&deg[dst[e]], 1.0f);
}

__global__ void dinv_kernel(float* __restrict__ deg, int n) {
    int i = blockIdx.x * blockDim.x + threadIdx.x;
    if (i < n) deg[i] = rsqrtf(deg[i] + 1.0f);   // +1 self loop; in-place -> dinv
}

// ---------------- WMMA GEMM: y[rows x 128] = X[rows x 128] @ W[128 x 128] ----------------
// If z != nullptr, row i of X is emb[z[i]] (fused embedding lookup).
__global__ __launch_bounds__(256) void gemm_kernel(const float* __restrict__ xin,
                                                   const int* __restrict__ z,
                                                   const float* __restrict__ emb,
                                                   const float* __restrict__ W,
                                                   float* __restrict__ y,
                                                   int nrows) {
    __shared__ float Ws[H * WPAD];
    __shared__ float Xs[16 * WPAD];
    const int tid = threadIdx.x;

    // Stage full W (128x128) into LDS, padded rows, float4 loads.
#pragma unroll
    for (int it = 0; it < 16; ++it) {
        int idx = (it * 256 + tid) * 4;          // 0..16383
        int r = idx >> 7, c = idx & 127;
        *(float4*)&Ws[r * WPAD + c] = *(const float4*)&W[idx];
    }

    // Stage 16-row X tile (with fused embedding gather for layer 1).
    const int rowbase = blockIdx.x * 16;
#pragma unroll
    for (int it = 0; it < 2; ++it) {
        int idx4 = it * 256 + tid;               // 512 float4 slots, 32 per row
        int r = idx4 >> 5;
        int c4 = (idx4 & 31) * 4;
        int gr = rowbase + r;
        if (gr >= nrows) gr = nrows - 1;         // clamp (harmless duplicate load)
        const float* srcrow = z ? (emb + (size_t)z[gr] * H) : (xin + (size_t)gr * H);
        *(float4*)&Xs[r * WPAD + c4] = *(const float4*)&srcrow[c4];
    }
    __syncthreads();

    const int wave  = tid >> 5;                  // 8 waves -> 8 column tiles of 16
    const int lane  = tid & 31;
    const int col0  = wave * 16;
    const int mrow  = lane & 15;                 // A: M index (both half-waves)
    const int khalf = (lane >> 4) * 2;           // A/B: K sub-offset (0 or 2)

    v8f acc = {};                                // 16x16 f32 accumulator (8 VGPRs)
#pragma unroll
    for (int k0 = 0; k0 < H; k0 += 4) {
        v2f a, b;
        // A 16x4 f32: lanes 0-15 hold K=k0+0/1, lanes 16-31 hold K=k0+2/3, row = lane&15
        a.x = Xs[mrow * WPAD + k0 + khalf + 0];
        a.y = Xs[mrow * WPAD + k0 + khalf + 1];
        // B 4x16 f32: row = K, col striped over lanes (same half-wave K split)
        b.x = Ws[(k0 + khalf + 0) * WPAD + col0 + mrow];
        b.y = Ws[(k0 + khalf + 1) * WPAD + col0 + mrow];
        acc = __builtin_amdgcn_wmma_f32_16x16x4_f32(false, a, false, b,
                                                    (short)0, acc, false, false);
    }

    // D layout: lanes 0-15 -> M=0..7 (VGPR r), lanes 16-31 -> M=8..15, N = lane&15
    const int mbase = (lane >> 4) * 8;
    const int ncol  = col0 + (lane & 15);
#pragma unroll
    for (int r = 0; r < 8; ++r) {
        int grow = rowbase + mbase + r;
        if (grow < nrows) y[(size_t)grow * H + ncol] = acc[r];
    }
}

// ---------------- edge scatter: agg[dst] += y[src] * dinv[src]*dinv[dst] ----------------
__global__ void scatter_kernel(const int* __restrict__ src, const int* __restrict__ dst,
                               const float* __restrict__ dinv,
                               const float* __restrict__ y, float* __restrict__ agg, int E) {
    int gid = blockIdx.x * blockDim.x + threadIdx.x;   // E * 32 threads
    int e = gid >> 5;
    if (e >= E) return;
    int c4 = (gid & 31) << 2;
    int s = src[e], d = dst[e];
    float norm = dinv[s] * dinv[d];
    float4 v = *(const float4*)&y[(size_t)s * H + c4];
    float* ap = &agg[(size_t)d * H + c4];
    atomicAdd(ap + 0, v.x * norm);
    atomicAdd(ap + 1, v.y * norm);
    atomicAdd(ap + 2, v.z * norm);
    atomicAdd(ap + 3, v.w * norm);
}

// ---------------- finalize: agg = relu(agg + y*dinv^2 + b) (in place) ----------------
__global__ void finalize_kernel(float* __restrict__ agg, const float* __restrict__ y,
                                const float* __restrict__ dinv, const float* __restrict__ b,
                                int total) {
    int i = blockIdx.x * blockDim.x + threadIdx.x;
    if (i >= total) return;
    int n = i >> 7, h = i & 127;
    float di = dinv[n];
    float v = agg[i] + y[i] * di * di + b[h];
    agg[i] = v > 0.0f ? v : 0.0f;
}

// ---------------- pooling ----------------
__global__ void pool_kernel(const float* __restrict__ x, const int* __restrict__ batch,
                            float* __restrict__ sums, int n) {
    int gid = blockIdx.x * blockDim.x + threadIdx.x;   // n * 32 threads
    int node = gid >> 5;
    if (node >= n) return;
    int c4 = (gid & 31) << 2;
    int g = batch[node];
    float4 v = *(const float4*)&x[(size_t)node * H + c4];
    float* sp = &sums[(size_t)g * H + c4];
    atomicAdd(sp + 0, v.x);
    atomicAdd(sp + 1, v.y);
    atomicAdd(sp + 2, v.z);
    atomicAdd(sp + 3, v.w);
}

__global__ void count_kernel(const int* __restrict__ batch, float* __restrict__ cnts, int n) {
    int i = blockIdx.x * blockDim.x + threadIdx.x;
    if (i < n) atomicAdd(&cnts[batch[i]], 1.0f);
}

// ---------------- head: silu(pooled @ Wr1 + br1) @ Wr2 + br2 ----------------
__global__ __launch_bounds__(64) void head_kernel(const float* __restrict__ sums,
                                                  const float* __restrict__ cnts,
                                                  const float* __restrict__ Wr1,
                                                  const float* __restrict__ br1,
                                                  const float* __restrict__ Wr2,
                                                  const float* __restrict__ br2,
                                                  float* __restrict__ out) {
    __shared__ float pooled[H];
    __shared__ float hred[64];
    const int g = blockIdx.x;
    const int j = threadIdx.x;                 // 0..63
    float inv = 1.0f / fmaxf(cnts[g], 1.0f);
    pooled[j]      = sums[(size_t)g * H + j]      * inv;
    pooled[j + 64] = sums[(size_t)g * H + j + 64] * inv;
    __syncthreads();
    float acc = br1[j];
#pragma unroll 8
    for (int k = 0; k < H; ++k) acc += pooled[k] * Wr1[k * 64 + j];
    float sil = acc / (1.0f + __expf(-acc));   // silu
    hred[j] = sil * Wr2[j];
    __syncthreads();
    for (int s = 32; s > 0; s >>= 1) {
        if (j < s) hred[j] += hred[j + s];
        __syncthreads();
    }
    if (j == 0) out[g] = hred[0] + br2[0];
}

static inline size_t align_up(size_t x, size_t a) { return (x + a - 1) & ~(a - 1); }

extern "C" void kernel_launch(void* const* d_in, const int* in_sizes, int n_in,
                              void* d_out, int out_size, void* d_ws, size_t ws_size,
                              hipStream_t stream) {
    // inputs: pos(0) z(1) batch(2) edge_index(3) emb(4) W1(5) b1(6) W2(7) b2(8)
    //         Wr1(9) br1(10) Wr2(11) br2(12)
    const int*   z     = (const int*)d_in[1];
    const int*   batch = (const int*)d_in[2];
    const int*   eidx  = (const int*)d_in[3];
    const float* emb   = (const float*)d_in[4];
    const float* W1    = (const float*)d_in[5];
    const float* b1    = (const float*)d_in[6];
    const float* W2    = (const float*)d_in[7];
    const float* b2    = (const float*)d_in[8];
    const float* Wr1   = (const float*)d_in[9];
    const float* br1   = (const float*)d_in[10];
    const float* Wr2   = (const float*)d_in[11];
    const float* br2   = (const float*)d_in[12];
    float* out = (float*)d_out;

    const int N = in_sizes[1];
    const int E = in_sizes[3] / 2;
    const int G = out_size;
    const int* esrc = eidx;
    const int* edst = eidx + E;
    const size_t NH = (size_t)N * H;

    // workspace carve-up
    char* w = (char*)d_ws;
    size_t off = 0;
    float* dinv = (float*)(w + off); off += align_up((size_t)N * 4, 256);
    float* buf0 = (float*)(w + off); off += align_up(NH * 4, 256);        // y = x @ W
    float* buf1 = (float*)(w + off); off += align_up(NH * 4, 256);        // agg / layer output
    float* sums = (float*)(w + off); off += align_up((size_t)G * H * 4, 256);
    float* cnts = (float*)(w + off); off += align_up((size_t)G * 4, 256);
    (void)ws_size;

    const int gemm_blocks  = (N + 15) / 16;
    const int scat_blocks  = (int)(((size_t)E * 32 + 255) / 256);
    const int fin_blocks   = (int)((NH + 255) / 256);
    const int pool_blocks  = (int)(((size_t)N * 32 + 255) / 256);
    const int node_blocks  = (N + 255) / 256;
    const int edge_blocks  = (E + 255) / 256;

    // degrees -> dinv (in place)
    hipMemsetAsync(dinv, 0, (size_t)N * 4, stream);
    deg_kernel<<<edge_blocks, 256, 0, stream>>>(edst, dinv, E);
    dinv_kernel<<<node_blocks, 256, 0, stream>>>(dinv, N);

    // ---- layer 1 (embedding fused into GEMM A-load) ----
    gemm_kernel<<<gemm_blocks, 256, 0, stream>>>(nullptr, z, emb, W1, buf0, N);
    hipMemsetAsync(buf1, 0, NH * 4, stream);
    scatter_kernel<<<scat_blocks, 256, 0, stream>>>(esrc, edst, dinv, buf0, buf1, E);
    finalize_kernel<<<fin_blocks, 256, 0, stream>>>(buf1, buf0, dinv, b1, (int)NH);

    // ---- layer 2 ----
    gemm_kernel<<<gemm_blocks, 256, 0, stream>>>(buf1, nullptr, emb, W2, buf0, N);
    hipMemsetAsync(buf1, 0, NH * 4, stream);
    scatter_kernel<<<scat_blocks, 256, 0, stream>>>(esrc, edst, dinv, buf0, buf1, E);
    finalize_kernel<<<fin_blocks, 256, 0, stream>>>(buf1, buf0, dinv, b2, (int)NH);

    // ---- global mean pool + head ----
    hipMemsetAsync(sums, 0, (size_t)G * H * 4, stream);
    hipMemsetAsync(cnts, 0, (size_t)G * 4, stream);
    pool_kernel<<<pool_blocks, 256, 0, stream>>>(buf1, batch, sums, N);
    count_kernel<<<node_blocks, 256, 0, stream>>>(batch, cnts, N);
    head_kernel<<<G, 64, 0, stream>>>(sums, cnts, Wr1, br1, Wr2, br2, out);
}